// GCNModel_34514357191054
// MI455X (gfx1250) — compile-verified
//
#include <hip/hip_runtime.h>

typedef float v2f __attribute__((ext_vector_type(2)));
typedef float v8f __attribute__((ext_vector_type(8)));

#define N_NODES  100000
#define N_EDGES  3200000
#define IN_DIM   512
#define HID      64
#define OUT_DIM  5

// Non-returning device-scope fp32 atomic add (guaranteed hardware atomic,
// no CAS-loop expansion). STOREcnt is drained by S_ENDPGM's implicit wait.
__device__ __forceinline__ void global_fadd(float* p, float v) {
    asm volatile("global_atomic_add_f32 %0, %1, off scope:SCOPE_DEV"
                 :: "v"(p), "v"(v) : "memory");
}

// ---------------------------------------------------------------- utilities
__global__ void fill_zero_kernel(float* __restrict__ p, long n) {
    long i = (long)blockIdx.x * blockDim.x + threadIdx.x;
    if (i < n) p[i] = 0.0f;
}

__global__ void degree_kernel(const int* __restrict__ dst,
                              const float* __restrict__ ew,
                              float* __restrict__ deg, long E) {
    long e = (long)blockIdx.x * blockDim.x + threadIdx.x;
    if (e >= E) return;
    global_fadd(&deg[dst[e]], ew[e]);
}

__global__ void dinv_kernel(const float* __restrict__ deg,
                            float* __restrict__ dinv, int M) {
    int i = blockIdx.x * blockDim.x + threadIdx.x;
    if (i < M) dinv[i] = rsqrtf(deg[i] + 1.0f);   // self-loop weight 1
}

// ------------------------------------------------ WMMA GEMM: C[M,64] = A[M,K] @ W[K,64]
// fp32 exact via V_WMMA_F32_16X16X4_F32. One wave computes a 16x64 output tile.
// W is staged in LDS k-pair interleaved so each B fragment is a single
// aligned ds_load_b64 (no VGPR re-pairing moves), spread over all 64 banks.
template <int K>
__global__ __launch_bounds__(256) void gemm_wmma_n64(
        const float* __restrict__ A, const float* __restrict__ W,
        float* __restrict__ C, int M) {
    constexpr int CK = 64;                  // K-chunk staged in LDS
    constexpr int PR = 160;                 // floats per k-pair row (128 used + 32-bank skew)
    __shared__ float Wlds[(CK / 2) * PR];   // 20 KB

    const int tid     = threadIdx.x;
    const int lane    = tid & 31;
    const int wave    = tid >> 5;
    const int laneRow = lane & 15;          // M (A) / N (B,C) position
    const int laneHi  = lane >> 4;          // K-half select

    const long r0 = ((long)blockIdx.x * 8 + wave) * 16;
    long row = r0 + laneRow;
    if (row >= M) row = M - 1;              // clamp: EXEC must stay all-1s for WMMA
    const float* arow = A + row * (long)K;

    v8f acc0 = {}, acc1 = {}, acc2 = {}, acc3 = {};

    for (int kb = 0; kb < K; kb += CK) {
        // cooperative swizzled stage of W[kb:kb+CK, 0:64]:
        //   Wlds[(k>>1)*PR + 2*n + (k&1)] = W[kb+k][n]
        const float4* wsrc = (const float4*)(W + (long)kb * 64);
        #pragma unroll
        for (int i = 0; i < (CK * 16) / 256; ++i) {
            const int idx = tid + i * 256;          // float4 index
            const int k   = idx >> 4;
            const int n0  = (idx & 15) << 2;
            const float4 w = wsrc[idx];
            float* p = &Wlds[(k >> 1) * PR + 2 * n0 + (k & 1)];
            p[0] = w.x; p[2] = w.y; p[4] = w.z; p[6] = w.w;
        }
        __syncthreads();

        if (kb + CK < K)                            // global_prefetch_b8 of next A chunk
            __builtin_prefetch(arow + kb + CK + 2 * laneHi, 0, 0);

        #pragma unroll
        for (int kk = 0; kk < CK; kk += 4) {
            // A frag: lanes 0-15 hold K=kk..kk+1, lanes 16-31 hold K=kk+2..kk+3
            v2f a = *(const v2f*)(arow + kb + kk + 2 * laneHi);
            // B frag pair {W[kq][n], W[kq+1][n]}, kq = kk + 2*laneHi, contiguous in LDS
            const float* wb = &Wlds[(kk / 2 + laneHi) * PR + 2 * laneRow];
            v2f b0 = *(const v2f*)(wb +  0);        // n = laneRow
            v2f b1 = *(const v2f*)(wb + 32);        // n = 16 + laneRow
            v2f b2 = *(const v2f*)(wb + 64);        // n = 32 + laneRow
            v2f b3 = *(const v2f*)(wb + 96);        // n = 48 + laneRow
            acc0 = __builtin_amdgcn_wmma_f32_16x16x4_f32(false, a, false, b0,
                                                         (short)0, acc0, false, false);
            acc1 = __builtin_amdgcn_wmma_f32_16x16x4_f32(false, a, false, b1,
                                                         (short)0, acc1, false, false);
            acc2 = __builtin_amdgcn_wmma_f32_16x16x4_f32(false, a, false, b2,
                                                         (short)0, acc2, false, false);
            acc3 = __builtin_amdgcn_wmma_f32_16x16x4_f32(false, a, false, b3,
                                                         (short)0, acc3, false, false);
        }
        __syncthreads();
    }

    // C layout: VGPR i -> rows r0+i (lanes 0-15) and r0+i+8 (lanes 16-31)
    #pragma unroll
    for (int i = 0; i < 8; ++i) {
        const long orow = r0 + i + 8 * laneHi;
        if (orow < M) {
            float* crow = C + orow * 64 + laneRow;
            crow[0]  = acc0[i];
            crow[16] = acc1[i];
            crow[32] = acc2[i];
            crow[48] = acc3[i];
        }
    }
}

// --------------------------------------------- edge gather + scatter-add (one wave/edge)
__global__ __launch_bounds__(256) void edge_agg_kernel(
        const float* __restrict__ h,
        const int* __restrict__ src,
        const int* __restrict__ dst,
        const float* __restrict__ ew,
        const float* __restrict__ dinv,
        float* __restrict__ agg, long E) {
    // wave-uniform edge id (readfirstlane -> SMEM loads for edge metadata)
    const int wv = __builtin_amdgcn_readfirstlane((int)(threadIdx.x >> 5));
    const long e = (long)blockIdx.x * 8 + wv;
    if (e >= E) return;                      // uniform; E % 8 == 0 so never taken
    const int lane = threadIdx.x & 31;
    const int s = src[e];
    const int d = dst[e];
    const float norm = dinv[s] * ew[e] * dinv[d];

    const int f = lane << 1;                 // 32 lanes x float2 = 64 feats (256B)
    const float2 hv = *(const float2*)(h + (long)s * 64 + f);
    float* ap = agg + (long)d * 64 + f;
    global_fadd(ap,     norm * hv.x);
    global_fadd(ap + 1, norm * hv.y);
}

// ------------------ out = relu(agg + dinv^2 * h + b); also re-zero agg for next layer
__global__ void finalize_kernel(float* __restrict__ agg,
                                const float* __restrict__ h,
                                const float* __restrict__ dinv,
                                const float* __restrict__ b,
                                float* __restrict__ out, int M) {
    long t = (long)blockIdx.x * blockDim.x + threadIdx.x;
    if (t >= (long)M * 64) return;
    const int n = (int)(t >> 6);
    const int f = (int)(t & 63);
    const float di = dinv[n];
    const float v = agg[t] + di * di * h[t] + b[f];
    agg[t] = 0.0f;                           // fused clear for next accumulation
    out[t] = fmaxf(v, 0.0f);
}

// ------------------------------------------------ head: out[M,5] = h[M,64] @ Wf + bf
__global__ void final_linear_kernel(const float* __restrict__ h,
                                    const float* __restrict__ Wf,
                                    const float* __restrict__ bf,
                                    float* __restrict__ out, int M) {
    long t = (long)blockIdx.x * blockDim.x + threadIdx.x;
    if (t >= (long)M * OUT_DIM) return;
    const int n = (int)(t / OUT_DIM);
    const int o = (int)(t % OUT_DIM);
    const float* hr = h + (long)n * 64;
    float s = bf[o];
    #pragma unroll
    for (int f = 0; f < 64; ++f) s = fmaf(hr[f], Wf[f * OUT_DIM + o], s);
    out[t] = s;
}

// ---------------------------------------------------------------------------- launch
extern "C" void kernel_launch(void* const* d_in, const int* in_sizes, int n_in,
                              void* d_out, int out_size, void* d_ws, size_t ws_size,
                              hipStream_t stream) {
    const float* x  = (const float*)d_in[0];
    const int*   ei = (const int*)  d_in[1];
    const float* ea = (const float*)d_in[2];
    const float* W1 = (const float*)d_in[3];
    const float* b1 = (const float*)d_in[4];
    const float* W2 = (const float*)d_in[5];
    const float* b2 = (const float*)d_in[6];
    const float* Wf = (const float*)d_in[7];
    const float* bf = (const float*)d_in[8];
    float* out = (float*)d_out;

    const int* src = ei;
    const int* dst = ei + N_EDGES;

    float* ws   = (float*)d_ws;
    float* h    = ws;                   // [100000, 64]
    float* agg  = ws + 6400000;         // [100000, 64]
    float* o1   = ws + 12800000;        // [100000, 64]
    float* dinv = ws + 19200000;        // [100000]
    float* deg  = ws + 19300000;        // [100000]

    const dim3 B(256);
    const int gNode  = (N_NODES + 255) / 256;            // 391
    const int gEdge  = (N_EDGES + 255) / 256;            // 12500
    const int gFeat  = (N_NODES * 64 + 255) / 256;       // 25000
    const int gEdgeW = N_EDGES / 8;                      // 400000 (8 edges/block)
    const int gGemm  = ((N_NODES + 15) / 16 + 7) / 8;    // 782
    const int gHead  = (N_NODES * OUT_DIM + 255) / 256;  // 1954

    // ---- normalization: deg = segsum(ew, dst); dinv = rsqrt(deg + 1)
    fill_zero_kernel<<<gNode, B, 0, stream>>>(deg, N_NODES);
    degree_kernel<<<gEdge, B, 0, stream>>>(dst, ea, deg, N_EDGES);
    dinv_kernel<<<gNode, B, 0, stream>>>(deg, dinv, N_NODES);

    // ---- layer 1
    gemm_wmma_n64<IN_DIM><<<gGemm, B, 0, stream>>>(x, W1, h, N_NODES);
    fill_zero_kernel<<<gFeat, B, 0, stream>>>(agg, (long)N_NODES * 64);
    edge_agg_kernel<<<gEdgeW, B, 0, stream>>>(h, src, dst, ea, dinv, agg, N_EDGES);
    finalize_kernel<<<gFeat, B, 0, stream>>>(agg, h, dinv, b1, o1, N_NODES);

    // ---- layer 2 (agg was re-zeroed by finalize_kernel)
    gemm_wmma_n64<HID><<<gGemm, B, 0, stream>>>(o1, W2, h, N_NODES);
    edge_agg_kernel<<<gEdgeW, B, 0, stream>>>(h, src, dst, ea, dinv, agg, N_EDGES);
    finalize_kernel<<<gFeat, B, 0, stream>>>(agg, h, dinv, b2, o1, N_NODES);

    // ---- head
    final_linear_kernel<<<gHead, B, 0, stream>>>(o1, Wf, bf, out, N_NODES);
}